// CritiGraph_35579509080217
// MI455X (gfx1250) — compile-verified
//
#include <hip/hip_runtime.h>
#include <stdint.h>

// CritiGraph update step for MI455X (gfx1250, wave32).
// H=16 bits, TP=8 dims, K=32, M=1025 candidates, BS=256, NB=64.
#define HB   16
#define TPD  8
#define KC   32
#define NBH  64
#define MJ   512      // H*K  (index of the "ori" column before permutation)
#define MM   1025     // 2*H*K + 1

// ---------------- CDNA5 async global->LDS helpers (ASYNCcnt path) ----------
typedef int v4i __attribute__((ext_vector_type(4)));
typedef __attribute__((address_space(1))) v4i* global_v4i_ptr;
typedef __attribute__((address_space(3))) v4i* lds_v4i_ptr;

__device__ __forceinline__ void async_g2l_b128(const void* gp, void* lp) {
#if __has_builtin(__builtin_amdgcn_global_load_async_to_lds_b128)
  __builtin_amdgcn_global_load_async_to_lds_b128(
      (global_v4i_ptr)(uintptr_t)gp,
      (lds_v4i_ptr)(unsigned)(uintptr_t)lp,   // LDS ptrs are 32-bit offsets
      0, 0);
#else
  unsigned           l32 = (unsigned)(uintptr_t)lp;          // LDS byte offset
  unsigned long long g64 = (unsigned long long)(uintptr_t)gp;
  asm volatile("global_load_async_to_lds_b128 %0, %1, off"
               :: "v"(l32), "v"(g64) : "memory");
#endif
}

__device__ __forceinline__ void wait_async0() {
#if __has_builtin(__builtin_amdgcn_s_wait_asynccnt)
  __builtin_amdgcn_s_wait_asynccnt(0);
#else
  asm volatile("s_wait_asynccnt 0" ::: "memory");
#endif
}

// Decode candidate |c| for permuted index value mp at coordinate t.
// cnc order before perm: [res(512), ori(1), -res(512)], res = (|sta|^ (1<<h)) ^ rm.
__device__ __forceinline__ int decode_abs(int mp, int t, const int* mlow,
                                          const int* staAbs, int* negOut) {
  if (mp == MJ) { *negOut = 0; return staAbs[t]; }
  int j = mp, neg = 0;
  if (mp > MJ) { j = mp - (MJ + 1); neg = 1; }
  int rm = mlow[(j * TPD + t) * 2];           // low dword of int64 mask elem
  *negOut = neg;
  return (staAbs[t] ^ (1 << (j >> 5))) ^ rm;  // h = j / K = j >> 5
}

__global__ __launch_bounds__(256)
void critigraph_main(const long long* __restrict__ locations,
                     const long long* __restrict__ sta_ind,
                     const long long* __restrict__ pos_ind,
                     const float*     __restrict__ logits,
                     const long long* __restrict__ rmasks,
                     const long long* __restrict__ perm,
                     const long long* __restrict__ lg,
                     const unsigned char* __restrict__ maskb,
                     float* __restrict__ outLoc,   // d_out+1: new_locations (f32)
                     float* __restrict__ wsPart)   // [BS] partial loss sums
{
  __shared__ __align__(16) long long sMask[MJ * TPD];  // 32 KB rmask slice (async)
  __shared__ __align__(16) long long sPos[NBH * TPD];  //  4 KB gathered pos rows
  __shared__ float4 sPPB[NBH * TPD];  // (|p| bits, mf*sign(p)/128, B, 0)
  __shared__ int    sStaAbs[TPD];
  __shared__ float  sStaSgn[TPD];
  __shared__ float  sRedV[256];
  __shared__ int    sRedI[256];
  __shared__ float  sMinT[TPD];

  const int tid = threadIdx.x;
  const int b   = blockIdx.x;
  const long long sidx = sta_ind[b];

  // (1) async-stage this block's 32KB random_masks slice: 2048 x 16B chunks
  const long long* gm = rmasks + (size_t)b * (MJ * TPD);
  #pragma unroll
  for (int it = 0; it < 8; ++it) {
    int c16 = tid + it * 256;
    async_g2l_b128(gm + c16 * 2, &sMask[c16 * 2]);
  }
  // (2) async-gather 64 pos rows (64B each = 4 x 16B chunks)
  {
    int n = tid >> 2, c = tid & 3;
    long long row = pos_ind[(size_t)b * NBH + n];
    async_g2l_b128(locations + row * TPD + c * 2, &sPos[n * TPD + c * 2]);
  }
  // (3) sta row (direct, overlaps with outstanding async copies)
  if (tid < TPD) {
    long long s = locations[sidx * TPD + tid];
    sStaAbs[tid] = (int)(s < 0 ? -s : s);
    sStaSgn[tid] = (s > 0) ? 1.f : (s < 0 ? -1.f : 0.f);
  }
  wait_async0();
  __syncthreads();

  // (4) per-(n,t): |p|, sign(p), dis_sta_pos.  1-table[x] = (clz(x+1)-16)/16.
  for (int i = tid; i < NBH * TPD; i += 256) {
    int t = i & 7;
    long long p = sPos[i];
    int   pa = (int)(p < 0 ? -p : p);
    float ps = (p > 0) ? 1.f : (p < 0 ? -1.f : 0.f);
    int   xr = sStaAbs[t] ^ pa;
    float d  = sStaSgn[t] * ps * (float)(__clz(xr + 1) - 16) * 0.0625f;
    sPPB[i] = make_float4(__int_as_float(pa), ps, d, 0.f);
  }
  __syncthreads();

  // (5) per-n: S = sum_t d; fold mask & logit:
  //     B[n,t] = mf*((S-d)/8 - logit),  y[n,t] = mf*sign(p)/128
  if (tid < NBH) {
    int n = tid;
    float S = 0.f;
    #pragma unroll
    for (int t = 0; t < TPD; ++t) S += sPPB[n * TPD + t].z;
    float lz = logits[(size_t)b * NBH + n];
    float mf = maskb[(size_t)b * NBH + n] ? 1.f : 0.f;
    #pragma unroll
    for (int t = 0; t < TPD; ++t) {
      float4 v = sPPB[n * TPD + t];
      float B = mf * fmaf(S - v.z, 0.125f, -lz);
      float y = mf * v.y * (1.f / 128.f);
      sPPB[n * TPD + t] = make_float4(v.x, y, B, 0.f);
    }
  }
  __syncthreads();

  // (6) candidate sweep: thread (q,t) owns m = q, q+32, ... for coordinate t.
  const int* mlow = (const int*)sMask;
  const int t = tid & 7, q = tid >> 3;
  float bestV = 3.402823e38f;
  int   bestM = 0;
  for (int m = q; m < MM; m += 32) {
    int mp = (int)perm[m];
    int neg;
    int   cabs = decode_abs(mp, t, mlow, sStaAbs, &neg);
    float csgn = (cabs == 0) ? 0.f : (neg ? -1.f : 1.f);
    float acc = 0.f;
    #pragma unroll 16
    for (int n = 0; n < NBH; ++n) {
      float4 v = sPPB[n * TPD + t];                  // ds_load_b128, bank-free
      int   xr = cabs ^ __float_as_int(v.x);
      float g  = (float)(__clz(xr + 1) - 16);        // (1 - table[xor]) * 16
      float val = fmaf(csgn * v.y, g, v.z);          // dc/8 + B
      acc += fabsf(val);
    }
    if (acc < bestV) { bestV = acc; bestM = m; }     // strict < keeps first m
  }

  // (7) argmin reduction across the 32 q-threads sharing each t (tie -> min m)
  sRedV[tid] = bestV; sRedI[tid] = bestM;
  __syncthreads();
  #pragma unroll
  for (int s = 128; s >= 8; s >>= 1) {
    if (tid < s) {
      float v2 = sRedV[tid + s]; int i2 = sRedI[tid + s];
      if (v2 < sRedV[tid] || (v2 == sRedV[tid] && i2 < sRedI[tid])) {
        sRedV[tid] = v2; sRedI[tid] = i2;
      }
    }
    __syncthreads();
  }

  // (8) scatter winning codes; per-block loss partial
  if (tid < TPD) {
    int mp = (int)perm[sRedI[tid]];
    int neg;
    int cabs = decode_abs(mp, tid, mlow, sStaAbs, &neg);
    int cs   = neg ? -cabs : cabs;
    outLoc[sidx * TPD + tid] = (float)cs;            // exact: |c| < 2^16
    sMinT[tid] = sRedV[tid] / (float)lg[b];
  }
  __syncthreads();
  if (tid == 0) {
    float s = 0.f;
    #pragma unroll
    for (int t2 = 0; t2 < TPD; ++t2) s += sMinT[t2];
    wsPart[b] = s;
  }
}

__global__ void copy_locations_f32(const long long* __restrict__ loc,
                                   float* __restrict__ out, int n) {
  for (int i = blockIdx.x * blockDim.x + threadIdx.x; i < n;
       i += gridDim.x * blockDim.x)
    out[i] = (float)loc[i];
}

__global__ void finalize_tl(const float* __restrict__ part,
                            float* __restrict__ out0, int bs, float scale) {
  __shared__ float sh[256];
  int tid = threadIdx.x;
  sh[tid] = (tid < bs) ? part[tid] : 0.f;
  __syncthreads();
  for (int s = 128; s > 0; s >>= 1) {
    if (tid < s) sh[tid] += sh[tid + s];
    __syncthreads();
  }
  if (tid == 0) out0[0] = sh[0] * scale;
}

extern "C" void kernel_launch(void* const* d_in, const int* in_sizes, int n_in,
                              void* d_out, int out_size, void* d_ws, size_t ws_size,
                              hipStream_t stream) {
  (void)n_in; (void)out_size; (void)ws_size;
  const long long*     locations = (const long long*)d_in[0];
  const long long*     sta_ind   = (const long long*)d_in[1];
  const long long*     pos_ind   = (const long long*)d_in[2];
  const float*         logits    = (const float*)    d_in[3];
  const long long*     rmasks    = (const long long*)d_in[4];
  const long long*     perm      = (const long long*)d_in[5];
  const long long*     lg        = (const long long*)d_in[6];
  const unsigned char* maskb     = (const unsigned char*)d_in[7];
  // d_in[8] = table: unused — computed analytically via V_CLZ on device.

  float* out = (float*)d_out;     // [0]=tl, [1..]=new_locations (f32-exact)
  float* wsf = (float*)d_ws;      // BS partial loss sums
  const int nLoc = in_sizes[0];   // VOCAB*TP
  const int bs   = in_sizes[1];   // 256

  copy_locations_f32<<<256, 256, 0, stream>>>(locations, out + 1, nLoc);
  critigraph_main<<<bs, 256, 0, stream>>>(locations, sta_ind, pos_ind, logits,
                                          rmasks, perm, lg, maskb, out + 1, wsf);
  finalize_tl<<<1, 256, 0, stream>>>(wsf, out, bs, 1.f / (float)(bs * TPD));
}